// BregMarginLoss_1812476199257
// MI455X (gfx1250) — compile-verified
//
#include <hip/hip_runtime.h>
#include <hip/hip_bf16.h>

typedef __attribute__((ext_vector_type(2))) float v2f;
typedef __attribute__((ext_vector_type(8))) float v8f;

#define B_ROWS 4096
#define DFEAT  256
#define NROWS  8192   // 2*B_ROWS

// ---------------------------------------------------------------------------
// Kernel 0: zero the 257-float output (loss + 256-bin histogram).
// ---------------------------------------------------------------------------
__global__ void breg_init_out(float* __restrict__ out) {
    int i = blockIdx.x * blockDim.x + threadIdx.x;
    if (i < 257) out[i] = 0.0f;
}

// ---------------------------------------------------------------------------
// Kernel 1: per-row max + argmax (first-index tie-break), histogram of argmax.
// One wave32 per row; 8 waves (256 threads) per block; 1024 blocks.
// Each lane loads 8 contiguous floats (2 x b128), local scan keeps lowest
// index on ties, then a shfl_xor butterfly reduces (max, min-index) over the
// 32 lanes (wave32!).
// ---------------------------------------------------------------------------
__global__ __launch_bounds__(256) void breg_rowmax(
    const float* __restrict__ a, const float* __restrict__ b,
    float* __restrict__ ws_max, int* __restrict__ ws_idx,
    float* __restrict__ out_hist) {

    const int wave = threadIdx.x >> 5;
    const int lane = threadIdx.x & 31;
    const int r    = blockIdx.x * 8 + wave;

    const float* row = (r < B_ROWS) ? (a + (size_t)r * DFEAT)
                                    : (b + (size_t)(r - B_ROWS) * DFEAT);
    const float4* row4 = (const float4*)row;

    float4 x = row4[lane * 2 + 0];   // cols [8*lane,   8*lane+4)
    float4 y = row4[lane * 2 + 1];   // cols [8*lane+4, 8*lane+8)

    const int base = lane * 8;
    float v[8] = {x.x, x.y, x.z, x.w, y.x, y.y, y.z, y.w};
    float vmax = v[0];
    int   imax = base;
#pragma unroll
    for (int k = 1; k < 8; ++k) {
        if (v[k] > vmax) { vmax = v[k]; imax = base + k; }   // strict > keeps first index
    }

    // Cross-lane all-reduce of (max, min index on tie) over 32 lanes.
#pragma unroll
    for (int off = 16; off > 0; off >>= 1) {
        float ov = __shfl_xor(vmax, off, 32);
        int   oi = __shfl_xor(imax, off, 32);
        if (ov > vmax || (ov == vmax && oi < imax)) { vmax = ov; imax = oi; }
    }

    if (lane == 0) {
        ws_max[r] = vmax;
        ws_idx[r] = imax;
        atomicAdd(&out_hist[imax], 1.0f);   // counts are integers -> exact in f32
    }
}

// ---------------------------------------------------------------------------
// Kernel 2: per-row loss terms + reduction.
//   pos_r = max_feat[r] - features[r, idx_max[pair(r)]]
//   neg_r = max_feat[r] - features[r, idx_max[c(r)]]
//   term  = pos_r + relu(1 - neg_r)^2
// Wave-level sum via V_WMMA_F32_16X16X4_F32 with B = ones:
//   A[m,0]=term[lane m], A[m,2]=term[lane m+16], other K zero
//   => D[m,n] = term[m] + term[m+16] for every column n.
// Sum of 8 D components + shfl_xor(.,16) = full 32-term wave sum.
// EXEC is all ones here (8192 = 32 blocks * 256 threads, no divergence).
// ---------------------------------------------------------------------------
__global__ __launch_bounds__(256) void breg_loss(
    const float* __restrict__ a, const float* __restrict__ b,
    const int* __restrict__ neg_choice,
    const float* __restrict__ ws_max, const int* __restrict__ ws_idx,
    float* __restrict__ out_loss) {

    const int r  = blockIdx.x * 256 + threadIdx.x;
    const int pr = (r < B_ROWS) ? (r + B_ROWS) : (r - B_ROWS);

    const float* row = (r < B_ROWS) ? (a + (size_t)r * DFEAT)
                                    : (b + (size_t)(r - B_ROWS) * DFEAT);

    const float mf  = ws_max[r];
    const float pos = mf - row[ws_idx[pr]];

    const int e1 = (r < pr) ? r : pr;
    const int e2 = (r < pr) ? pr : r;
    int c = neg_choice[r];
    c += (c >= e1) ? 1 : 0;
    c += (c >= e2) ? 1 : 0;            // sequential skip, matches reference
    const float neg = mf - row[ws_idx[c]];

    const float m    = fmaxf(1.0f - neg, 0.0f);
    const float term = pos + m * m;

    // ---- wave32 sum via WMMA f32 16x16x4 (sum-by-ones) ----
    v2f av; av[0] = term; av[1] = 0.0f;   // A tile: K=0 (lanes 0-15) / K=2 (lanes 16-31)
    v2f bv; bv[0] = 1.0f; bv[1] = 1.0f;   // B tile: all ones
    v8f cv = {};
    v8f dv = __builtin_amdgcn_wmma_f32_16x16x4_f32(
        /*neg_a=*/false, av, /*neg_b=*/false, bv,
        /*c_mod=*/(short)0, cv, /*reuse_a=*/false, /*reuse_b=*/false);

    float partial = 0.0f;
#pragma unroll
    for (int i = 0; i < 8; ++i) partial += dv[i];
    const float wsum = partial + __shfl_xor(partial, 16, 32);

    __shared__ float sred[8];
    if ((threadIdx.x & 31) == 0) sred[threadIdx.x >> 5] = wsum;
    __syncthreads();
    if (threadIdx.x == 0) {
        float s = 0.0f;
#pragma unroll
        for (int i = 0; i < 8; ++i) s += sred[i];
        atomicAdd(out_loss, 0.5f * s);   // reference divides the total by 2
    }
}

// ---------------------------------------------------------------------------
extern "C" void kernel_launch(void* const* d_in, const int* in_sizes, int n_in,
                              void* d_out, int out_size, void* d_ws, size_t ws_size,
                              hipStream_t stream) {
    const float* a   = (const float*)d_in[0];   // out_a [4096,256] f32
    const float* b   = (const float*)d_in[1];   // out_b [4096,256] f32
    const int*   nc  = (const int*)d_in[2];     // neg_choice [8192] i32
    float*       out = (float*)d_out;           // [0]=loss, [1..257)=num_max

    float* ws_max = (float*)d_ws;                                  // 8192 f32
    int*   ws_idx = (int*)((char*)d_ws + NROWS * sizeof(float));   // 8192 i32

    breg_init_out<<<2, 256, 0, stream>>>(out);
    breg_rowmax <<<NROWS / 8, 256, 0, stream>>>(a, b, ws_max, ws_idx, out + 1);
    breg_loss   <<<NROWS / 256, 256, 0, stream>>>(a, b, nc, ws_max, ws_idx, out);
}